// Multiple_GIOU_loss_16071767622176
// MI455X (gfx1250) — compile-verified
//
#include <hip/hip_runtime.h>
#include <cstdint>

#define NUM_CLASSES 80
#define BATCH 32
#define NGT   256
#define NPRED 2048
#define CHUNK 256              // predictions per block (1 per thread)
#define NCHUNKS (NPRED / CHUNK)
#define NWAVES (CHUNK / 32)

// ---------------------------------------------------------------------------
// Kernel 1: one block per (batch, chunk-of-256-preds). 256 threads (8 wave32).
// Stages the batch's 256 GT boxes + labels into LDS via gfx1250 async
// global->LDS loads, precomputes GT-side invariants (area, +1-shifted coords,
// packed label) once per box, builds the per-batch label histogram in LDS,
// computes per-prediction GIoU loss, reduces via wave32 shuffles.
// ---------------------------------------------------------------------------
__global__ __launch_bounds__(256) void giou_loss_kernel(
    const float* __restrict__ imgs_box,    // [B, G, 4]
    const int*   __restrict__ img_labels,  // [B, G]
    const float* __restrict__ pre_box,     // [B, P, 4]
    const int*   __restrict__ pre_label,   // [B, P]
    float*       __restrict__ partial)     // [B * NCHUNKS]
{
    __shared__ float4 s_gt[NGT];           // 4 KB GT boxes (x1,y1,x2+1,y2+1)
    __shared__ int    s_glab[NGT];         // 1 KB GT labels (async landing pad)
    __shared__ float2 s_ta_lab[NGT];       // 2 KB {tarea, label-bits}
    __shared__ int    s_cnt[NUM_CLASSES];  // per-batch label histogram
    __shared__ float  s_wsum[NWAVES];      // per-wave partial sums

    const int tid   = threadIdx.x;
    const int chunk = blockIdx.x;
    const int b     = blockIdx.y;

    // ---- Stage GT tile into LDS with CDNA5 async global->LDS loads --------
    // Low 32 bits of a generic pointer into the LDS aperture == LDS byte
    // offset (ISA 10.2).
    {
        const float4* gsrc = reinterpret_cast<const float4*>(imgs_box) + (size_t)b * NGT + tid;
        unsigned lds_box = (unsigned)(uintptr_t)(&s_gt[tid]);
        asm volatile("global_load_async_to_lds_b128 %0, %1, off"
                     :: "v"(lds_box), "v"(gsrc) : "memory");

        const int* lsrc = img_labels + (size_t)b * NGT + tid;
        unsigned lds_lab = (unsigned)(uintptr_t)(&s_glab[tid]);
        asm volatile("global_load_async_to_lds_b32 %0, %1, off"
                     :: "v"(lds_lab), "v"(lsrc) : "memory");
    }

    if (tid < NUM_CLASSES) s_cnt[tid] = 0;

    // Wait for this wave's async copies, then barrier so all 8 waves' LDS
    // writes (and the histogram zeroing) are visible block-wide.
    asm volatile("s_wait_asynccnt 0x0" ::: "memory");
    __syncthreads();

    // ---- Pre-pass: per-GT invariants + per-batch label histogram ---------
    {
        // Each thread owns GT box `tid` (read/modify own slot only: no hazard).
        const float4 tb = s_gt[tid];
        const float tw = fmaxf(tb.z - tb.x + 1.0f, 0.0f);
        const float th = fmaxf(tb.w - tb.y + 1.0f, 0.0f);
        s_ta_lab[tid] = make_float2(tw * th, __int_as_float(s_glab[tid]));
        s_gt[tid] = make_float4(tb.x, tb.y, tb.z + 1.0f, tb.w + 1.0f);
    }
    const int* lab_b = pre_label + (size_t)b * NPRED;
    #pragma unroll
    for (int i = 0; i < NPRED / CHUNK; ++i) {
        atomicAdd(&s_cnt[lab_b[i * CHUNK + tid]], 1);   // deterministic int adds
    }
    __syncthreads();

    // ---- This thread's prediction ----------------------------------------
    const int p = chunk * CHUNK + tid;
    const float4 pb = reinterpret_cast<const float4*>(pre_box)[(size_t)b * NPRED + p];
    const int pl = lab_b[p];

    const float px2p  = pb.z + 1.0f;        // x2 + 1
    const float py2p  = pb.w + 1.0f;        // y2 + 1
    const float pw    = fmaxf(pb.z - pb.x + 1.0f, 0.0f);
    const float ph    = fmaxf(pb.w - pb.y + 1.0f, 0.0f);
    const float parea = pw * ph;
    const float inv_parea = 1.0f / parea;   // loop-invariant: hoist the IEEE div

    float sel_giou = 0.0f;
    int   last_idx = -1;

    #pragma unroll 4
    for (int g = 0; g < NGT; ++g) {
        const float4 tb = s_gt[g];          // (x1, y1, x2+1, y2+1)
        const float2 m  = s_ta_lab[g];      // (tarea, label-bits)

        const float iw = fmaxf(fminf(px2p, tb.z) - fmaxf(pb.x, tb.x), 0.0f);
        const float ih = fmaxf(fminf(py2p, tb.w) - fmaxf(pb.y, tb.y), 0.0f);
        const float inter = iw * ih;
        const float uni   = parea + m.x - inter;

        // NOTE: reference divides by pred_area (not union) — keep that.
        const float iou = fmaxf(inter * inv_parea, 1e-6f);

        const float ow = fmaxf(fmaxf(px2p, tb.z) - fminf(pb.x, tb.x), 0.0f);
        const float oh = fmaxf(fmaxf(py2p, tb.w) - fminf(pb.y, tb.y), 0.0f);
        const float outer = ow * oh;

        // v_rcp_f32 (~1 ULP) instead of the full IEEE div sequence.
        float giou = iou - (outer - uni) * __builtin_amdgcn_rcpf(outer);
        giou = fminf(fmaxf(giou, -1.0f), 1.0f);   // clip before validity test

        const bool valid = (pl == __float_as_int(m.y)) & (giou > 0.0f);
        if (valid) { last_idx = g; sel_giou = giou; }  // ascending g -> last wins
    }

    const float loss_p = (last_idx >= 0) ? (1.0f - sel_giou) : 0.0f;
    const float weight = (float)s_cnt[pl];

    // ---- Wave32 shuffle reduction (lane-permute path, no barriers) -------
    float v = weight * loss_p;
    #pragma unroll
    for (int off = 16; off > 0; off >>= 1) v += __shfl_down(v, off, 32);
    if ((tid & 31) == 0) s_wsum[tid >> 5] = v;
    __syncthreads();
    if (tid == 0) {
        float t = 0.0f;
        #pragma unroll
        for (int w = 0; w < NWAVES; ++w) t += s_wsum[w];
        partial[b * NCHUNKS + chunk] = t;
    }
}

// ---------------------------------------------------------------------------
// Kernel 2: reduce the 256 partials -> scalar, scale by 1/B.
// ---------------------------------------------------------------------------
__global__ __launch_bounds__(256) void giou_reduce_kernel(
    const float* __restrict__ partial, float* __restrict__ out)
{
    __shared__ float s_wsum[NWAVES];
    const int tid = threadIdx.x;
    float v = partial[tid];
    #pragma unroll
    for (int off = 16; off > 0; off >>= 1) v += __shfl_down(v, off, 32);
    if ((tid & 31) == 0) s_wsum[tid >> 5] = v;
    __syncthreads();
    if (tid == 0) {
        float t = 0.0f;
        #pragma unroll
        for (int w = 0; w < NWAVES; ++w) t += s_wsum[w];
        out[0] = t * (1.0f / (float)BATCH);
    }
}

extern "C" void kernel_launch(void* const* d_in, const int* in_sizes, int n_in,
                              void* d_out, int out_size, void* d_ws, size_t ws_size,
                              hipStream_t stream) {
    const float* imgs_box   = (const float*)d_in[0];  // [32,256,4] f32
    const int*   img_labels = (const int*)  d_in[1];  // [32,256]   i32
    const float* pre_box    = (const float*)d_in[2];  // [32,2048,4] f32
    const int*   pre_label  = (const int*)  d_in[3];  // [32,2048]  i32
    float* out = (float*)d_out;
    float* partial = (float*)d_ws;                    // 256 floats scratch

    dim3 grid(NCHUNKS, BATCH);                        // 8 x 32 = 256 blocks
    giou_loss_kernel<<<grid, CHUNK, 0, stream>>>(imgs_box, img_labels,
                                                 pre_box, pre_label, partial);
    giou_reduce_kernel<<<1, 256, 0, stream>>>(partial, out);
}